// SlidingWindowAttention_17884243821070
// MI455X (gfx1250) — compile-verified
//
#include <hip/hip_runtime.h>
#include <hip/hip_bf16.h>
#include <math.h>

typedef __attribute__((ext_vector_type(2))) float v2f;
typedef __attribute__((ext_vector_type(8))) float v8f;

#define EMBED    1024
#define HEADS    16
#define HEAD_DIM 64
#define SEQ      2048
#define BATCH    2
#define HALF_WIN 64
#define KTILES   9            // 144-key window per 16-query tile
#define KWIN     (KTILES*16)  // 144

__device__ __forceinline__ v8f wmma_f32(v2f a, v2f b, v8f c) {
  // V_WMMA_F32_16X16X4_F32 : D = A(16x4,f32) * B(4x16,f32) + C(16x16,f32)
  return __builtin_amdgcn_wmma_f32_16x16x4_f32(false, a, false, b, (short)0, c,
                                               false, false);
}

// 32 contiguous bytes global -> LDS via async DMA (ASYNCcnt-tracked).
// INST_OFFSET is added to BOTH the LDS and global addresses (ISA 08 §4.4),
// so one (lds, gaddr) pair covers 4 x b64. 8B alignment required (holds for
// the stride-18-word padded LDS tiles: word offset is always even).
__device__ __forceinline__ void async_copy32B(unsigned lds, unsigned long long g) {
  asm volatile(
      "global_load_async_to_lds_b64 %0, %1, off\n\t"
      "global_load_async_to_lds_b64 %0, %1, off offset:8\n\t"
      "global_load_async_to_lds_b64 %0, %1, off offset:16\n\t"
      "global_load_async_to_lds_b64 %0, %1, off offset:24"
      :: "v"(lds), "v"(g) : "memory");
}

__device__ __forceinline__ void wait_async0() {
  asm volatile("s_wait_asynccnt 0x0" ::: "memory");
}

// ---------------------------------------------------------------------------
// Kernel 1: sliding-window attention for one 16-query tile per wave32.
// ctx[b][q][h*64+d] = softmax_band(QK^T/32) @ V
// Out-of-range / out-of-band keys: addresses are CLAMPED (always-safe loads,
// no exec-mask branches); their scores are masked to -inf and their V rows are
// multiplied by p == 0, so loaded garbage never reaches the output.
// ---------------------------------------------------------------------------
__global__ __launch_bounds__(32)
void swa_attn_kernel(const float* __restrict__ Vg, const float* __restrict__ Kg,
                     const float* __restrict__ Qg, float* __restrict__ ctx)
{
  __shared__ float P[16][KWIN];            // probabilities, C-layout -> A-layout relay

  const int lane = threadIdx.x & 31;
  const int lo   = lane & 15;              // N (and M for A/B fragments)
  const int off  = (lane >> 4) ? 2 : 0;    // K sub-offset for A/B fragments
  const int hi8  = (lane >> 4) ? 8 : 0;    // row offset for C/D fragments

  const int bid = blockIdx.x;
  const int q0  = (bid & 127) * 16;        // SEQ/16 = 128 tiles
  const int h   = (bid >> 7) & 15;
  const int b   = bid >> 11;

  const size_t bS   = (size_t)b * SEQ;
  const size_t hoff = (size_t)h * HEAD_DIM;

  // ---- preload Q A-fragments: row q0+lo, dims {4i+off, 4i+off+1} ----
  v2f qa[16];
  {
    const float* qrow = Qg + (bS + q0 + lo) * EMBED + hoff;
    #pragma unroll
    for (int i = 0; i < 16; ++i)
      qa[i] = *(const v2f*)(qrow + 4 * i + off);
  }

  // ---- scores: 9 key tiles of 16x16, K-dim = 64 in chunks of 4 ----
  v8f sc[KTILES];
  #pragma unroll
  for (int j = 0; j < KTILES; ++j) {
    int kr = q0 - HALF_WIN + 16 * j + lo;  // key row for this lane's N
    kr = (kr < 0) ? 0 : ((kr > SEQ - 1) ? SEQ - 1 : kr);   // clamp, mask later
    const float* krow = Kg + (bS + kr) * EMBED + hoff;
    v2f kb[16];
    #pragma unroll
    for (int i = 0; i < 16; ++i)
      kb[i] = *(const v2f*)(krow + 4 * i + off);
    v8f c = {};
    #pragma unroll
    for (int i = 0; i < 16; ++i)
      c = wmma_f32(qa[i], kb[i], c);
    sc[j] = c;
  }

  // ---- mask + softmax (reference: energy->-1e20 outside band, then /32) ----
  const float scale = 0.03125f;            // 1/sqrt(1024)
  float rmax[8], rinv[8];
  #pragma unroll
  for (int r = 0; r < 8; ++r) {
    const int m = r + hi8;                 // query row in tile (C-layout)
    float mx = -3.0e38f;
    #pragma unroll
    for (int j = 0; j < KTILES; ++j) {
      const int  k     = q0 - HALF_WIN + 16 * j + lo;
      const int  d     = k - (q0 + m);
      const bool valid = (k >= 0) && (k < SEQ) && (d >= -HALF_WIN) && (d <= HALF_WIN);
      const float sv   = valid ? sc[j][r] * scale : -3.0e38f;
      sc[j][r] = sv;
      mx = fmaxf(mx, sv);
    }
    #pragma unroll
    for (int sh = 8; sh >= 1; sh >>= 1)    // reduce across the 16-lane half
      mx = fmaxf(mx, __shfl_xor(mx, sh, 32));
    rmax[r] = mx;
  }
  #pragma unroll
  for (int r = 0; r < 8; ++r) {
    float s = 0.f;
    #pragma unroll
    for (int j = 0; j < KTILES; ++j) {
      const float sv = sc[j][r];
      const float p  = (sv > -1.0e37f) ? __expf(sv - rmax[r]) : 0.f;
      s += p;
      P[r + hi8][16 * j + lo] = p;         // relay to A-layout via LDS
    }
    #pragma unroll
    for (int sh = 8; sh >= 1; sh >>= 1)
      s += __shfl_xor(s, sh, 32);
    rinv[r] = 1.0f / s;
  }

  // ---- out = P(16x144) @ V(144x64), 4 column tiles ----
  #pragma unroll
  for (int t = 0; t < 4; ++t) {
    v8f acc = {};
    #pragma unroll 4
    for (int c = 0; c < KWIN / 4; ++c) {   // 36 K-chunks
      const int c0 = 4 * c;
      const v2f a  = *(const v2f*)&P[lo][c0 + off];
      int r0 = q0 - HALF_WIN + c0 + off;   // rows feeding b.x / b.y
      int r1 = r0 + 1;
      r0 = (r0 < 0) ? 0 : ((r0 > SEQ - 1) ? SEQ - 1 : r0);  // clamp: p==0 there
      r1 = (r1 < 0) ? 0 : ((r1 > SEQ - 1) ? SEQ - 1 : r1);
      v2f bv;
      bv.x = Vg[(bS + r0) * EMBED + hoff + t * 16 + lo];
      bv.y = Vg[(bS + r1) * EMBED + hoff + t * 16 + lo];
      acc = wmma_f32(a, bv, acc);
    }
    #pragma unroll
    for (int r = 0; r < 8; ++r)
      ctx[(bS + q0 + r + hi8) * EMBED + hoff + t * 16 + lo] = acc[r] * rinv[r];
  }
}

// ---------------------------------------------------------------------------
// Kernel 2: out = ctx @ w_out^T + b_out   (M=4096, N=1024, K=1024)
// 64x64 block tile, 4 waves. LDS staging via GLOBAL_LOAD_ASYNC_TO_LDS_B64
// (ASYNCcnt), stride-18-word padding -> conflict-free, 8B-aligned fragments.
// ---------------------------------------------------------------------------
__global__ __launch_bounds__(128)
void swa_proj_kernel(const float* __restrict__ X, const float* __restrict__ W,
                     const float* __restrict__ bias, float* __restrict__ out)
{
  __shared__ float As[64][18];
  __shared__ float Bs[64][18];

  const int tid  = threadIdx.x;
  const int wave = tid >> 5;
  const int lane = tid & 31;
  const int lo   = lane & 15;
  const int off  = (lane >> 4) ? 2 : 0;
  const int hi8  = (lane >> 4) ? 8 : 0;

  const int m0 = blockIdx.x * 64;
  const int n0 = blockIdx.y * 64;

  v8f acc[4] = {{}, {}, {}, {}};

  const int srow = tid >> 1;               // 64 rows x 2 threads
  const int scol = (tid & 1) * 8;          // 8 floats (32B) per thread per tile
  const unsigned ldsA = (unsigned)(size_t)&As[srow][scol];
  const unsigned ldsB = (unsigned)(size_t)&Bs[srow][scol];
  const float* xrow = X + (size_t)(m0 + srow) * EMBED + scol;
  const float* wrow = W + (size_t)(n0 + srow) * EMBED + scol;

  for (int k0 = 0; k0 < EMBED; k0 += 16) {
    // async DMA: global -> LDS, no VGPR round trip
    async_copy32B(ldsA, (unsigned long long)(size_t)(xrow + k0));
    async_copy32B(ldsB, (unsigned long long)(size_t)(wrow + k0));
    if (k0 + 16 < EMBED) {                 // warm L2 for next tile
      __builtin_prefetch(xrow + k0 + 16, 0, 0);
      __builtin_prefetch(wrow + k0 + 16, 0, 0);
    }
    wait_async0();
    __syncthreads();

    #pragma unroll
    for (int c = 0; c < 4; ++c) {
      const v2f a = *(const v2f*)&As[wave * 16 + lo][4 * c + off];
      #pragma unroll
      for (int t = 0; t < 4; ++t) {
        const v2f bv = *(const v2f*)&Bs[t * 16 + lo][4 * c + off];
        acc[t] = wmma_f32(a, bv, acc[t]);
      }
    }
    __syncthreads();
  }

  #pragma unroll
  for (int t = 0; t < 4; ++t) {
    const float bb = bias[n0 + t * 16 + lo];
    #pragma unroll
    for (int r = 0; r < 8; ++r)
      out[(size_t)(m0 + wave * 16 + r + hi8) * EMBED + n0 + t * 16 + lo] =
          acc[t][r] + bb;
  }
}

// ---------------------------------------------------------------------------
extern "C" void kernel_launch(void* const* d_in, const int* in_sizes, int n_in,
                              void* d_out, int out_size, void* d_ws, size_t ws_size,
                              hipStream_t stream) {
  // setup_inputs order: values, keys, queries, mask, w_out, b_out
  const float* Vg   = (const float*)d_in[0];
  const float* Kg   = (const float*)d_in[1];
  const float* Qg   = (const float*)d_in[2];
  const float* W    = (const float*)d_in[4];
  const float* bias = (const float*)d_in[5];
  float*       out  = (float*)d_out;
  float*       ctx  = (float*)d_ws;   // needs BATCH*SEQ*EMBED*4 = 16 MiB scratch

  // attention: one wave32 per 16-query tile per (batch, head)
  dim3 agrid(BATCH * HEADS * (SEQ / 16));
  swa_attn_kernel<<<agrid, 32, 0, stream>>>(Vg, Kg, Qg, ctx);

  // projection: 64x64 output tiles
  dim3 pgrid((BATCH * SEQ) / 64, EMBED / 64);
  swa_proj_kernel<<<pgrid, 128, 0, stream>>>(ctx, W, bias, out);
}